// EdgeDecoder_10050223472991
// MI455X (gfx1250) — compile-verified
//
#include <hip/hip_runtime.h>
#include <hip/hip_bf16.h>

typedef __attribute__((ext_vector_type(2))) float v2f;
typedef __attribute__((ext_vector_type(8))) float v8f;

#define H 128

// OUT[N, 128] = X[N,128] @ W^T + b   (torch Linear semantics; W is [128,128] row-major)
// One wave per 16x16 output tile using V_WMMA_F32_16X16X4_F32 (fp32 matrix path).
// Block = 256 threads = 8 waves; wave w handles output-feature tile j0 = 16*w,
// block m handles row tile m0 = 16*blockIdx.x. K loop: 128 / 4 = 32 WMMAs.
__global__ void linear_wmma_kernel(const float* __restrict__ X,
                                   const float* __restrict__ W,
                                   const float* __restrict__ bias,
                                   float* __restrict__ OUT,
                                   int N) {
    const int wave = threadIdx.x >> 5;   // 0..7 -> which 16-wide j tile
    const int lane = threadIdx.x & 31;
    const int lmod = lane & 15;
    const int lhi  = lane >> 4;          // 0 or 1
    const int m0 = blockIdx.x * 16;
    const int j0 = wave * 16;

    // A fragment rows: lane holds M = lmod; guard tail by clamping the load row.
    int mrow = m0 + lmod;
    if (mrow > N - 1) mrow = N - 1;
    const float* __restrict__ xrow = X + (size_t)mrow * H;
    // B fragment: B[k, n] = W[j0 + n, k]; lane holds N = lmod.
    const float* __restrict__ wrow = W + (size_t)(j0 + lmod) * H;

    v8f acc = {};
#pragma unroll
    for (int k0 = 0; k0 < H; k0 += 4) {
        const int k = k0 + 2 * lhi;
        // ISA layout (32-bit A 16x4): VGPR0 = A[m, k0+2*lhi], VGPR1 = A[m, k0+2*lhi+1]
        v2f a = *(const v2f*)(xrow + k);
        v2f b = *(const v2f*)(wrow + k);
        // 8 args: (neg_a, A, neg_b, B, c_mod, C, reuse_a, reuse_b)
        acc = __builtin_amdgcn_wmma_f32_16x16x4_f32(false, a, false, b,
                                                    (short)0, acc, false, false);
    }

    // C/D layout: lane L, VGPR v -> m = v + 8*(L/16), n = L%16 (n uniform over v).
    const float bj = bias[j0 + lmod];
    const int mbase = m0 + 8 * lhi;
    float* __restrict__ orow = OUT + (size_t)mbase * H + (j0 + lmod);
#pragma unroll
    for (int v = 0; v < 8; ++v) {
        if (mbase + v < N) {
            orow[(size_t)v * H] = acc[v] + bj;
        }
    }
}

// One wave32 per edge: lane l loads float4 (b128) slices of U[row] and R[col],
// partial dot, then 5-step wave32 xor-shuffle reduction. U/R live in the 192MB L2.
__global__ void edge_dot_kernel(const long long* __restrict__ rows,
                                const long long* __restrict__ cols,
                                const float* __restrict__ U,
                                const float* __restrict__ R,
                                float* __restrict__ out,
                                int nEdges) {
    const int wid  = (int)((blockIdx.x * (size_t)blockDim.x + threadIdx.x) >> 5);
    const int lane = threadIdx.x & 31;
    if (wid >= nEdges) return;

    const long long r = rows[wid];
    const long long c = cols[wid];

    const float4 a = ((const float4*)(U + (size_t)r * H))[lane];
    const float4 b = ((const float4*)(R + (size_t)c * H))[lane];

    float s = a.x * b.x + a.y * b.y + a.z * b.z + a.w * b.w;
#pragma unroll
    for (int off = 16; off > 0; off >>= 1) {
        s += __shfl_xor(s, off, 32);
    }
    if (lane == 0) {
        out[wid] = s;
    }
}

extern "C" void kernel_launch(void* const* d_in, const int* in_sizes, int n_in,
                              void* d_out, int out_size, void* d_ws, size_t ws_size,
                              hipStream_t stream) {
    const float*     z_user = (const float*)d_in[0];
    const float*     z_repo = (const float*)d_in[1];
    const long long* eli    = (const long long*)d_in[2];  // [2, E] int64
    const float*     user_W = (const float*)d_in[3];
    const float*     user_b = (const float*)d_in[4];
    const float*     repo_W = (const float*)d_in[5];
    const float*     repo_b = (const float*)d_in[6];
    float*           out    = (float*)d_out;

    const int nUser = in_sizes[0] / H;
    const int nRepo = in_sizes[1] / H;
    const int nEdge = in_sizes[2] / 2;

    // Workspace: U [nUser,128] then R [nRepo,128] (fp32) — 76.8 MB, L2-resident.
    float* U = (float*)d_ws;
    float* R = U + (size_t)nUser * H;

    // Phase 1: hoisted linear layers via fp32 WMMA.
    linear_wmma_kernel<<<(nUser + 15) / 16, 256, 0, stream>>>(z_user, user_W, user_b, U, nUser);
    linear_wmma_kernel<<<(nRepo + 15) / 16, 256, 0, stream>>>(z_repo, repo_W, repo_b, R, nRepo);

    // Phase 2: gather + dot, one wave per edge (8 waves / 256-thread block).
    const size_t totalThreads = (size_t)nEdge * 32;
    const int blocks = (int)((totalThreads + 255) / 256);
    edge_dot_kernel<<<blocks, 256, 0, stream>>>(eli, eli + nEdge, U, R, out, nEdge);
}